// ParallelRetention_reduce_79508434583787
// MI455X (gfx1250) — compile-verified
//
#include <hip/hip_runtime.h>

// ParallelRetention fused kernel for MI455X (gfx1250, wave32, WMMA).
//   O = rowscale( tril(Qbar Kbar^T) V ),  P_i = rowsum of causal scores.
// fp32 matrix path: V_WMMA_F32_16X16X4_F32 (exact precision, fp32 accumulate).
// omask (268 MB) streamed once with non-temporal loads; K/V stay L2-resident.

typedef float v2f __attribute__((ext_vector_type(2)));
typedef float v4f __attribute__((ext_vector_type(4)));
typedef float v8f __attribute__((ext_vector_type(8)));

#define Bc 2
#define Hc 8
#define Sc 2048
#define Dc 64
#define BHc (Bc * Hc)
#define BQ 64     // query rows per workgroup
#define KS 68     // LDS row stride (floats) for K and S tiles: conflict-free b64 reads
#define VS 72     // LDS row stride (floats) for V: half-waves hit disjoint bank halves

__launch_bounds__(128)
__global__ void retention_fused_kernel(const float* __restrict__ q,
                                       const float* __restrict__ k,
                                       const float* __restrict__ v,
                                       const float* __restrict__ omask,
                                       float* __restrict__ out)
{
    // heavy blocks (large qb) first for better tail scheduling
    const int qb   = (int)(gridDim.x - 1u - blockIdx.x);   // 0..31
    const int bh   = (int)blockIdx.y;                      // 0..15
    const int tid  = (int)threadIdx.x;                     // 0..127
    const int w    = tid >> 5;                             // wave 0..3
    const int lane = tid & 31;
    const int half = lane >> 4;                            // 0/1 (16-lane half)
    const int l16  = lane & 15;

    __shared__ float ldsK[BQ * KS];   // k_bar block, row = key j, col = d
    __shared__ float ldsV[BQ * VS];   // v block
    __shared__ float ldsS[BQ * KS];   // staged score strip (C-layout -> A-layout)
    __shared__ float ldsQC[BQ];       // q_bar coefficients for this block's rows

    const size_t bh_sd = (size_t)bh * Sc * Dc;
    const size_t bh_ss = (size_t)bh * Sc * (size_t)Sc;

    // ---- phase 1: rowsum(omask) for our 64 query rows (only consumer of it) ----
    // wave w owns rows [16w, 16w+16); 128-bit non-temporal stream, 16B aligned.
    for (int rr = 0; rr < 16; ++rr) {
        const int row = w * 16 + rr;
        const size_t base = bh_ss + (size_t)(qb * BQ + row) * Sc;
        const v4f* om4 = (const v4f*)(omask + base);
        float s = 0.f;
        for (int j = lane; j < Sc / 4; j += 32) {
            v4f t = __builtin_nontemporal_load(om4 + j);
            s += (t.x + t.y) + (t.z + t.w);
        }
        #pragma unroll
        for (int off = 16; off > 0; off >>= 1) s += __shfl_xor(s, off, 32);
        if (lane == 0) {
            float m0 = omask[base];            // omask[i][0]
            ldsQC[row] = m0 / sqrtf(s);        // q_bar coefficient
        }
    }
    __syncthreads();

    // ---- load Q A-fragments (16x16x4 f32 layout: lane=M%16, half selects K pair) ----
    const int qrow = qb * BQ + w * 16 + l16;
    const float qc = ldsQC[w * 16 + l16];
    const float* qptr = q + bh_sd + (size_t)qrow * Dc;
    v2f qf[16];
    #pragma unroll
    for (int kk = 0; kk < 16; ++kk) {
        const int c = kk * 4 + half * 2;
        v2f a; a.x = qptr[c] * qc; a.y = qptr[c + 1] * qc;
        qf[kk] = a;
    }

    v8f oacc[4];
    v8f pacc;
    #pragma unroll
    for (int r = 0; r < 8; ++r) pacc[r] = 0.f;
    #pragma unroll
    for (int ct = 0; ct < 4; ++ct) {
        #pragma unroll
        for (int r = 0; r < 8; ++r) oacc[ct][r] = 0.f;
    }

    // ---- flash loop over key blocks (causal: kb <= qb) ----
    for (int kb = 0; kb <= qb; ++kb) {
        __syncthreads();   // protect previous iteration's LDS reads
        {
            const float* kbase = k + bh_sd + (size_t)(kb * BQ) * Dc;
            const float* vbase = v + bh_sd + (size_t)(kb * BQ) * Dc;
            for (int idx = tid; idx < BQ * Dc / 4; idx += 128) {
                const int row  = idx >> 4;
                const int col4 = (idx & 15) * 4;
                const float kc = 1.f / omask[bh_ss + (size_t)(kb * BQ + row) * Sc];
                const v4f kvv = *(const v4f*)(kbase + row * Dc + col4);
                ldsK[row * KS + col4 + 0] = kvv.x * kc;
                ldsK[row * KS + col4 + 1] = kvv.y * kc;
                ldsK[row * KS + col4 + 2] = kvv.z * kc;
                ldsK[row * KS + col4 + 3] = kvv.w * kc;
                const v4f vvv = *(const v4f*)(vbase + row * Dc + col4);
                ldsV[row * VS + col4 + 0] = vvv.x;
                ldsV[row * VS + col4 + 1] = vvv.y;
                ldsV[row * VS + col4 + 2] = vvv.z;
                ldsV[row * VS + col4 + 3] = vvv.w;
            }
        }
        __syncthreads();

        // S strip = Q(16x64) * Kblk^T (4 tiles of 16x16), causal mask, P accum
        #pragma unroll
        for (int t = 0; t < 4; ++t) {
            v8f s;
            #pragma unroll
            for (int r = 0; r < 8; ++r) s[r] = 0.f;
            #pragma unroll
            for (int kk = 0; kk < 16; ++kk) {
                const int d0   = kk * 4 + half * 2;
                const int krow = t * 16 + l16;
                v2f b; b.x = ldsK[krow * KS + d0]; b.y = ldsK[krow * KS + d0 + 1];
                s = __builtin_amdgcn_wmma_f32_16x16x4_f32(
                        false, qf[kk], false, b, (short)0, s, false, false);
            }
            const int jg = kb * BQ + t * 16 + l16;      // global key index (col)
            #pragma unroll
            for (int r = 0; r < 8; ++r) {
                const int ig = qb * BQ + w * 16 + r + 8 * half;  // global query row
                const float sv = (jg <= ig) ? s[r] : 0.f;
                pacc[r] += sv;
                ldsS[(w * 16 + r + 8 * half) * KS + t * 16 + l16] = sv;
            }
        }
        __syncthreads();   // make staged S visible across lanes

        // refragment S strip into A-layout
        v2f sa[16];
        #pragma unroll
        for (int nn = 0; nn < 16; ++nn) {
            const int n0   = nn * 4 + half * 2;
            const int srow = w * 16 + l16;
            v2f a; a.x = ldsS[srow * KS + n0]; a.y = ldsS[srow * KS + n0 + 1];
            sa[nn] = a;
        }
        // O strip += S(16x64) * V(64x64)
        #pragma unroll
        for (int ct = 0; ct < 4; ++ct) {
            v8f o = oacc[ct];
            #pragma unroll
            for (int nn = 0; nn < 16; ++nn) {
                const int n0 = nn * 4 + half * 2;
                v2f b; b.x = ldsV[(n0    ) * VS + ct * 16 + l16];
                       b.y = ldsV[(n0 + 1) * VS + ct * 16 + l16];
                o = __builtin_amdgcn_wmma_f32_16x16x4_f32(
                        false, sa[nn], false, b, (short)0, o, false, false);
            }
            oacc[ct] = o;
        }
    }

    // ---- epilogue: P = max(|rowsum|, 1); O /= P; store ----
    float invp[8];
    #pragma unroll
    for (int r = 0; r < 8; ++r) {
        float p = pacc[r];
        p += __shfl_xor(p, 1, 32);
        p += __shfl_xor(p, 2, 32);
        p += __shfl_xor(p, 4, 32);
        p += __shfl_xor(p, 8, 32);           // reduce within 16-lane half
        invp[r] = 1.f / fmaxf(fabsf(p), 1.f);
    }
    #pragma unroll
    for (int ct = 0; ct < 4; ++ct) {
        #pragma unroll
        for (int r = 0; r < 8; ++r) {
            const int m = w * 16 + r + 8 * half;
            out[bh_sd + (size_t)(qb * BQ + m) * Dc + ct * 16 + l16]
                = oacc[ct][r] * invp[r];
        }
    }
}

extern "C" void kernel_launch(void* const* d_in, const int* in_sizes, int n_in,
                              void* d_out, int out_size, void* d_ws, size_t ws_size,
                              hipStream_t stream) {
    (void)in_sizes; (void)n_in; (void)out_size; (void)d_ws; (void)ws_size;
    const float* q  = (const float*)d_in[0];
    const float* k  = (const float*)d_in[1];
    const float* v  = (const float*)d_in[2];
    const float* om = (const float*)d_in[3];
    float* out = (float*)d_out;
    dim3 grid(Sc / BQ, BHc);   // (32, 16)
    retention_fused_kernel<<<grid, 128, 0, stream>>>(q, k, v, om, out);
}